// GeometricAttention_55808805045013
// MI455X (gfx1250) — compile-verified
//
#include <hip/hip_runtime.h>
#include <math.h>

typedef float v2f __attribute__((ext_vector_type(2)));
typedef float v8f __attribute__((ext_vector_type(8)));

#define DIMD   512
#define NHEADS 8
#define HD     64
#define BATCH  2
#define TLEN   2048
#define NTOK   (BATCH * TLEN)   // 4096

__device__ __forceinline__ v8f wmma_f32(v2f a, v2f b, v8f c) {
  // V_WMMA_F32_16X16X4_F32 : D[16x16] = A[16x4] * B[4x16] + C
  return __builtin_amdgcn_wmma_f32_16x16x4_f32(false, a, false, b, (short)0, c,
                                               false, false);
}

// ---------------------------------------------------------------------------
// Kernel 1: fused QKV projection.  y = x @ W.T + b for W in {Wq,Wk,Wv},
// scattered into planes laid out [B, H, T, HD] (f32) in workspace.
// One wave computes a 16x64 strip (4 accumulators) so each A (x) load is
// reused by 4 WMMAs; the 64-wide strip is head-aligned.
// ---------------------------------------------------------------------------
__global__ void __launch_bounds__(128) qkv_proj_kernel(
    const float* __restrict__ x,
    const float* __restrict__ Wq, const float* __restrict__ bq,
    const float* __restrict__ Wk, const float* __restrict__ bk,
    const float* __restrict__ Wv, const float* __restrict__ bv,
    float* __restrict__ qkv /* 3 planes of NTOK*DIMD floats */) {
  const int lane = threadIdx.x & 31;
  const int wave = blockIdx.x * (blockDim.x >> 5) + (threadIdx.x >> 5);
  const int NSTRIP = (3 * DIMD) / 64;       // 24 strips of 64 columns
  const int mt = wave / NSTRIP;
  const int st = wave % NSTRIP;
  const int m0 = mt * 16;
  const int ng = st * 64;                   // global column in [0,1536)
  const int which = ng / DIMD;              // 0=q 1=k 2=v
  const int n0 = ng % DIMD;                 // multiple of 64 -> single head

  const float* W    = (which == 0) ? Wq : (which == 1) ? Wk : Wv;
  const float* bias = (which == 0) ? bq : (which == 1) ? bk : bv;
  float* dst = qkv + (size_t)which * NTOK * DIMD;

  const int hi = lane >> 4, lo = lane & 15;

  v8f acc[4];
#pragma unroll
  for (int t = 0; t < 4; ++t) acc[t] = (v8f){0.f,0.f,0.f,0.f,0.f,0.f,0.f,0.f};

  const float* xrow = x + (size_t)(m0 + lo) * DIMD + 2 * hi; // A: x[m][k]
  const float* wrow = W + (size_t)(n0 + lo) * DIMD + 2 * hi; // B: W[n][k]
#pragma unroll 8
  for (int k = 0; k < DIMD; k += 4) {
    v2f a = *(const v2f*)(xrow + k);
#pragma unroll
    for (int t = 0; t < 4; ++t) {
      v2f b = *(const v2f*)(wrow + k + (size_t)t * 16 * DIMD);
      acc[t] = wmma_f32(a, b, acc[t]);
    }
  }

  const int h = n0 / HD;                    // constant for whole strip
#pragma unroll
  for (int t = 0; t < 4; ++t) {
    const float bv_ = bias[n0 + 16 * t + lo];
    const int d = 16 * t + lo;
#pragma unroll
    for (int r = 0; r < 8; ++r) {
      const int m = m0 + r + 8 * hi;        // token index
      const int bb = m >> 11, tok = m & (TLEN - 1);
      dst[(((size_t)bb * NHEADS + h) * TLEN + tok) * HD + d] = acc[t][r] + bv_;
    }
  }
}

// ---------------------------------------------------------------------------
// Kernel 2: flash-style geometric attention.
// One wave handles one (b, h, 16-query-row) tile, streaming 16-key tiles with
// online softmax.  Raw QK^T scores are transposed D->A layout through LDS;
// all elementwise math + softmax bookkeeping then happens in A layout where a
// lane owns half of one query row: row reductions are 1 shuffle, softmax state
// is one scalar per lane, and exp() directly produces the P.V A-operand.
// ---------------------------------------------------------------------------
__global__ void __launch_bounds__(32) geo_attn_kernel(
    const float* __restrict__ qkv, const float* __restrict__ curvature,
    float* __restrict__ attn_out /* [B, T, DIMD] */) {
  __shared__ float pl[16 * 16];             // score transpose bounce
  __shared__ float k2s[16];                 // ||k_j||^2, indexed by key-in-tile
  __shared__ float als[16];                 // per-row alpha (for O rescale)
  __shared__ float ls[16];                  // per-row l (final normalize)

  const int lane = threadIdx.x;
  const int hi = lane >> 4, lo = lane & 15;

  const int qt = blockIdx.x & 127;                  // query tile (T/16 = 128)
  const int h  = (blockIdx.x >> 7) & 7;
  const int b  = blockIdx.x >> 10;

  const size_t plane = (size_t)NTOK * DIMD;
  const size_t head  = (((size_t)b * NHEADS + h) * TLEN) * HD;
  const float* Q = qkv + head;
  const float* K = qkv + plane + head;
  const float* V = qkv + 2 * plane + head;
  const int i0 = qt * 16;

  const float c   = fabsf(curvature[h]) + 1e-8f;
  const float rsc = rsqrtf(c);

  // ---- preload Q tile (16 x 64) in A layout; q^2 of row `lo` per lane ----
  v2f qa[16];
  float q2p = 0.f;
  const float* qrow = Q + (size_t)(i0 + lo) * HD + 2 * hi;
#pragma unroll
  for (int ch = 0; ch < 16; ++ch) {
    v2f a = *(const v2f*)(qrow + 4 * ch);
    qa[ch] = a;
    q2p += a.x * a.x + a.y * a.y;
  }
  const float q2  = q2p + __shfl_xor(q2p, 16, 32);  // ||q_{i0+lo}||^2
  const float omq = 1.f - c * q2;

  float m = -INFINITY, l = 0.f;             // softmax state of row `lo`
  v8f o[4];
#pragma unroll
  for (int t = 0; t < 4; ++t) o[t] = (v8f){0.f,0.f,0.f,0.f,0.f,0.f,0.f,0.f};

  for (int j0 = 0; j0 < TLEN; j0 += 16) {
    // ---- S = Q * K^T (16 WMMA steps over hd=64), k^2 on the fly ----
    const float* krow = K + (size_t)(j0 + lo) * HD + 2 * hi;
    v8f s = {0.f, 0.f, 0.f, 0.f, 0.f, 0.f, 0.f, 0.f};
    float k2p = 0.f;
#pragma unroll
    for (int ch = 0; ch < 16; ++ch) {
      v2f kb = *(const v2f*)(krow + 4 * ch);
      k2p += kb.x * kb.x + kb.y * kb.y;
      s = wmma_f32(qa[ch], kb, s);
    }
    const float k2 = k2p + __shfl_xor(k2p, 16, 32);

    // ---- transpose raw scores D->A layout; publish k^2 ----
    __syncthreads();                        // prior iter's pl/k2s reads done
    if (hi == 0) k2s[lo] = k2;
#pragma unroll
    for (int r = 0; r < 8; ++r) pl[(r + 8 * hi) * 16 + lo] = s[r];
    __syncthreads();

    v2f sa[4], kc[4];
#pragma unroll
    for (int kk = 0; kk < 4; ++kk) {
      sa[kk] = *(const v2f*)(&pl[lo * 16 + 4 * kk + 2 * hi]);
      kc[kk] = *(const v2f*)(&k2s[4 * kk + 2 * hi]);
    }

    // ---- hyperbolic logits, A layout: lane owns half of row `lo` ----
    float sc[8];
#pragma unroll
    for (int kk = 0; kk < 4; ++kk) {
#pragma unroll
      for (int cpt = 0; cpt < 2; ++cpt) {
        const float qk  = cpt ? sa[kk].y : sa[kk].x;
        const float k2j = cpt ? kc[kk].y : kc[kk].x;
        const float num   = 2.f * (q2 + k2j - 2.f * qk);
        const float denom = fmaxf(omq * (1.f - c * k2j), 1e-5f);
        const float arg   = fmaxf(1.f + __fdividef(c * num, denom), 1.00001f);
        const float dist  =
            __logf(arg + __builtin_amdgcn_sqrtf(arg * arg - 1.f)) * rsc;
        sc[2 * kk + cpt] = -dist;           // TEMPERATURE = 1
      }
    }

    // ---- online softmax: 7 in-lane ops + one shuffle per reduction ----
    float tmax = sc[0];
#pragma unroll
    for (int e = 1; e < 8; ++e) tmax = fmaxf(tmax, sc[e]);
    tmax = fmaxf(tmax, __shfl_xor(tmax, 16, 32));
    const float mnew  = fmaxf(m, tmax);
    const float alpha = __expf(m - mnew);
    m = mnew;

    v2f pa[4];
    float rs = 0.f;
#pragma unroll
    for (int kk = 0; kk < 4; ++kk) {
      pa[kk].x = __expf(sc[2 * kk] - mnew);
      pa[kk].y = __expf(sc[2 * kk + 1] - mnew);
      rs += pa[kk].x + pa[kk].y;
    }
    rs += __shfl_xor(rs, 16, 32);
    l = l * alpha + rs;

    // ---- rescale O (needs alpha re-indexed to D-layout rows) ----
    if (hi == 0) als[lo] = alpha;
    __syncthreads();
#pragma unroll
    for (int t = 0; t < 4; ++t) {
#pragma unroll
      for (int r = 0; r < 8; ++r) o[t][r] *= als[r + 8 * hi];
    }

    // ---- O += P * V : pa[] is already the A operand ----
#pragma unroll
    for (int kk = 0; kk < 4; ++kk) {
      const float* vr = V + (size_t)(j0 + 4 * kk + 2 * hi) * HD;
#pragma unroll
      for (int hb = 0; hb < 4; ++hb) {
        v2f vb;
        vb.x = vr[hb * 16 + lo];
        vb.y = vr[HD + hb * 16 + lo];
        o[hb] = wmma_f32(pa[kk], vb, o[hb]);
      }
    }
  }

  // ---- normalize (l re-indexed to D-layout rows) and store ----
  __syncthreads();
  if (hi == 0) ls[lo] = l;
  __syncthreads();
  float linv[8];
#pragma unroll
  for (int r = 0; r < 8; ++r) linv[r] = 1.f / ls[r + 8 * hi];
#pragma unroll
  for (int hb = 0; hb < 4; ++hb) {
#pragma unroll
    for (int r = 0; r < 8; ++r) {
      const int i = i0 + r + 8 * hi;
      const int d = h * HD + hb * 16 + lo;
      attn_out[((size_t)b * TLEN + i) * DIMD + d] = o[hb][r] * linv[r];
    }
  }
}

// ---------------------------------------------------------------------------
// Kernel 3: output projection  out = attn @ Wo.T + bo  (16x64 strip per wave)
// ---------------------------------------------------------------------------
__global__ void __launch_bounds__(128) out_proj_kernel(
    const float* __restrict__ A, const float* __restrict__ Wo,
    const float* __restrict__ bo, float* __restrict__ Y) {
  const int lane = threadIdx.x & 31;
  const int wave = blockIdx.x * (blockDim.x >> 5) + (threadIdx.x >> 5);
  const int NSTRIP = DIMD / 64;             // 8
  const int mt = wave / NSTRIP, st = wave % NSTRIP;
  const int m0 = mt * 16, n0 = st * 64;
  const int hi = lane >> 4, lo = lane & 15;

  v8f acc[4];
#pragma unroll
  for (int t = 0; t < 4; ++t) acc[t] = (v8f){0.f,0.f,0.f,0.f,0.f,0.f,0.f,0.f};

  const float* ar = A  + (size_t)(m0 + lo) * DIMD + 2 * hi;
  const float* wr = Wo + (size_t)(n0 + lo) * DIMD + 2 * hi;
#pragma unroll 8
  for (int k = 0; k < DIMD; k += 4) {
    v2f a = *(const v2f*)(ar + k);
#pragma unroll
    for (int t = 0; t < 4; ++t) {
      v2f b = *(const v2f*)(wr + k + (size_t)t * 16 * DIMD);
      acc[t] = wmma_f32(a, b, acc[t]);
    }
  }
#pragma unroll
  for (int t = 0; t < 4; ++t) {
    const float bv_ = bo[n0 + 16 * t + lo];
#pragma unroll
    for (int r = 0; r < 8; ++r)
      Y[(size_t)(m0 + r + 8 * hi) * DIMD + n0 + 16 * t + lo] = acc[t][r] + bv_;
  }
}

// ---------------------------------------------------------------------------
extern "C" void kernel_launch(void* const* d_in, const int* in_sizes, int n_in,
                              void* d_out, int out_size, void* d_ws,
                              size_t ws_size, hipStream_t stream) {
  const float* x         = (const float*)d_in[0];
  const float* Wq        = (const float*)d_in[1];
  const float* bq        = (const float*)d_in[2];
  const float* Wk        = (const float*)d_in[3];
  const float* bk        = (const float*)d_in[4];
  const float* Wv        = (const float*)d_in[5];
  const float* bv        = (const float*)d_in[6];
  const float* Wo        = (const float*)d_in[7];
  const float* bo        = (const float*)d_in[8];
  const float* curvature = (const float*)d_in[9];

  float* ws   = (float*)d_ws;
  float* qkv  = ws;                                   // 3 planes, 24 MB
  float* attn = ws + 3 * (size_t)NTOK * DIMD;         // 8 MB

  // QKV projection: (4096/16) * (1536/64) = 6144 waves, 4 waves/block
  qkv_proj_kernel<<<6144 / 4, 128, 0, stream>>>(x, Wq, bq, Wk, bk, Wv, bv,
                                                qkv);
  // Attention: B*H*(T/16) = 2048 single-wave blocks
  geo_attn_kernel<<<BATCH * NHEADS * (TLEN / 16), 32, 0, stream>>>(
      qkv, curvature, attn);
  // Output projection: (4096/16) * (512/64) = 2048 waves, 4 waves/block
  out_proj_kernel<<<2048 / 4, 128, 0, stream>>>(attn, Wo, bo, (float*)d_out);
}